// QuadrupletInteraction_82660940578988
// MI455X (gfx1250) — compile-verified
//
#include <hip/hip_runtime.h>
#include <hip/hip_bf16.h>

typedef __attribute__((ext_vector_type(2))) float v2f;
typedef __attribute__((ext_vector_type(4))) float v4f;
typedef __attribute__((ext_vector_type(8))) float v8f;

#define KMAX   16
#define D_EDGE 128
#define D_SBF  32
#define D_QUAD 32
#define INV_SQRT_2 0.70710678118654752440f

__device__ __forceinline__ float silu(float v) {
    return v / (1.0f + __expf(-v));
}

// ---------------- pass 0/1: build deterministic inverse neighbor map ----------------
__global__ void k_init_inv(int* __restrict__ inv, int n) {
    int i = blockIdx.x * blockDim.x + threadIdx.x;
    if (i < n) inv[i] = -1;
}

__global__ void k_scatter_inv(const int* __restrict__ enb, const int* __restrict__ erag,
                              int* __restrict__ inv, int E, int E_NB) {
    int nIdx = blockIdx.x * blockDim.x + threadIdx.x;
    if (nIdx < E_NB) {
        int e = enb[nIdx], k = erag[nIdx];
        if ((unsigned)e < (unsigned)E && (unsigned)k < (unsigned)KMAX)
            inv[e * KMAX + k] = nIdx;
    }
}

// ---------------- pass 2: m = silu(m_st @ W_down)  (E x 128) @ (128 x 32) ----------------
// block = 128 thr = 4 waves; wave -> (rowTile half, col half); 32 rows / block.
// m_st is a 51MB one-shot stream -> non-temporal loads.
__global__ __launch_bounds__(128) void k_gemm_down(const float* __restrict__ A,
                                                   const float* __restrict__ Wd,
                                                   float* __restrict__ M, int E) {
    __shared__ float sW[D_EDGE * D_QUAD]; // 16 KB
    int tid = threadIdx.x;
    for (int i = tid; i < D_EDGE * D_QUAD; i += 128) sW[i] = Wd[i];
    __syncthreads();

    int wave = tid >> 5, lane = tid & 31;
    int row0 = (blockIdx.x * 2 + (wave >> 1)) * 16;
    int colBase = (wave & 1) * 16;
    int lr = lane & 15;
    int hi = lane >> 4;      // 0: K=0,1   1: K=2,3
    int ks = hi * 2;
    int r = row0 + lr; if (r > E - 1) r = E - 1;
    const float* arow = A + (size_t)r * D_EDGE + ks;
    int col = colBase + lr;

    v8f acc = {};
    #pragma unroll
    for (int k = 0; k < D_EDGE; k += 4) {
        v2f a = __builtin_nontemporal_load((const v2f*)(arow + k)); // A[r,k+ks], A[r,k+ks+1]
        v2f b; b.x = sW[(k + ks) * D_QUAD + col];
               b.y = sW[(k + ks + 1) * D_QUAD + col];
        acc = __builtin_amdgcn_wmma_f32_16x16x4_f32(false, a, false, b, (short)0, acc, false, false);
    }

    int mrow0 = row0 + hi * 8;
    #pragma unroll
    for (int j = 0; j < 8; ++j) {
        int rr = mrow0 + j;
        if (rr < E) M[(size_t)rr * D_QUAD + col] = silu(acc[j]);
    }
}

// ---------------- pass 3: sbar[e,s] = sum_k sbf[inv[e,k], s]  (dominant HBM pass) --------
// 4 edges per wave32, lane -> (edge = lane>>3, s-quad = lane&7); each k-step is one
// 512B global_load_b128 per wave. NT loads: 205MB one-shot stream, keep L2 for reused data.
__global__ __launch_bounds__(256) void k_reduce_sbf(const float* __restrict__ sbf,
                                                    const int* __restrict__ inv,
                                                    float* __restrict__ sbar, int E) {
    int tid = threadIdx.x;
    int wave = tid >> 5, lane = tid & 31;
    int sub = lane >> 3;           // edge within wave (0..3)
    int s4  = (lane & 7) * 4;      // s channel group
    int e = (blockIdx.x * 8 + wave) * 4 + sub;
    int ec = e < E ? e : E - 1;    // clamp for address formation only
    const int* ip = inv + (size_t)ec * KMAX;
    v4f acc = {0.f, 0.f, 0.f, 0.f};
    #pragma unroll
    for (int k = 0; k < KMAX; ++k) {
        int n = ip[k];
        if (n >= 0) {
            v4f v = __builtin_nontemporal_load((const v4f*)(sbf + (size_t)n * D_SBF + s4));
            acc += v;
        }
    }
    if (e < E) *(v4f*)(sbar + (size_t)e * D_SBF + s4) = acc;
}

// ---------------- pass 4: x = (sbar ⊗ m) @ W2, K = 1024, N = 32, scaled -----------------
// block = 256 thr = 8 waves; wave -> (rowTile = w>>1, colHalf = w&1); 64 rows / block.
// W2 in dynamic LDS (128 KB), pair-interleaved: idx = (kk>>1)*64 + o*2 + (kk&1).
__global__ __launch_bounds__(256) void k_bilinear(const float* __restrict__ Mbuf,
                                                  const float* __restrict__ sbarBuf,
                                                  const float* __restrict__ Wbil,
                                                  const float* __restrict__ scale_ptr,
                                                  float* __restrict__ X, int E) {
    extern __shared__ float sW2[]; // 32768 floats
    int tid = threadIdx.x;
    for (int d = tid; d < 1024 * 32; d += 256) {
        int pair = d & 1;
        int o = (d >> 1) & 31;
        int kk = ((d >> 6) << 1) + pair;    // kk = s*32 + q
        int s = kk >> 5, q = kk & 31;
        sW2[d] = Wbil[(q * 32 + s) * 32 + o];
    }
    __syncthreads();

    float scale = scale_ptr[0];
    int wave = tid >> 5, lane = tid & 31;
    int rowBase = blockIdx.x * 64 + (wave >> 1) * 16;
    int colBase = (wave & 1) * 16;
    int lr = lane & 15, hi = lane >> 4, ks = hi * 2;
    int r = rowBase + lr; if (r > E - 1) r = E - 1;

    float sr[32];                            // full sbar row (all s used by every lane)
    {
        const v4f* sp = (const v4f*)(sbarBuf + (size_t)r * D_SBF);
        #pragma unroll
        for (int i = 0; i < 8; ++i) {
            v4f v = sp[i];
            sr[4*i] = v.x; sr[4*i+1] = v.y; sr[4*i+2] = v.z; sr[4*i+3] = v.w;
        }
    }
    float mrl[16];                           // only this lane-half's K slice of m row
    {
        const float* mp = Mbuf + (size_t)r * D_QUAD + ks;
        #pragma unroll
        for (int t = 0; t < 8; ++t) {
            v2f v = *(const v2f*)(mp + 4 * t);
            mrl[2*t] = v.x; mrl[2*t+1] = v.y;
        }
    }
    int col = colBase + lr;
    v8f acc = {};
    #pragma unroll
    for (int s = 0; s < 32; ++s) {
        float t = sr[s];
        #pragma unroll
        for (int ti = 0; ti < 8; ++ti) {
            v2f a; a.x = t * mrl[2*ti]; a.y = t * mrl[2*ti+1];
            int kkl = s * 32 + 4 * ti + ks;                 // even
            v2f b = *(const v2f*)&sW2[(kkl >> 1) * 64 + col * 2];
            acc = __builtin_amdgcn_wmma_f32_16x16x4_f32(false, a, false, b, (short)0, acc, false, false);
        }
    }
    int xrow0 = rowBase + hi * 8;
    #pragma unroll
    for (int j = 0; j < 8; ++j) {
        int rr = xrow0 + j;
        if (rr < E) X[(size_t)rr * D_QUAD + col] = acc[j] * scale;
    }
}

// ---------------- pass 5: fused up-projections, K=32, N=128 (both weights) ---------------
// block = 256 thr = 8 waves; wave w -> col tile w; 16 rows / block.
__global__ __launch_bounds__(256) void k_gemm_up(const float* __restrict__ X,
                                                 const float* __restrict__ Wst,
                                                 const float* __restrict__ Wts,
                                                 float* __restrict__ Out,
                                                 float* __restrict__ Z, int E) {
    __shared__ float sWst[32 * 128];   // 16 KB, pair-interleaved
    __shared__ float sWts[32 * 128];   // 16 KB
    int tid = threadIdx.x;
    for (int d = tid; d < 32 * 128; d += 256) {
        int pair = d & 1;
        int c = (d >> 1) & 127;
        int k = ((d >> 8) << 1) + pair;
        sWst[d] = Wst[k * 128 + c];
        sWts[d] = Wts[k * 128 + c];
    }
    __syncthreads();

    int wave = tid >> 5, lane = tid & 31;
    int rowBase = blockIdx.x * 16;
    int colBase = wave * 16;
    int lr = lane & 15, hi = lane >> 4, ks = hi * 2;
    int r = rowBase + lr; if (r > E - 1) r = E - 1;

    float arl[16];
    {
        const float* xp = X + (size_t)r * D_QUAD + ks;
        #pragma unroll
        for (int t = 0; t < 8; ++t) {
            v2f v = *(const v2f*)(xp + 4 * t);
            arl[2*t] = v.x; arl[2*t+1] = v.y;
        }
    }
    int col = colBase + lr;
    v8f acc_st = {}, acc_ts = {};
    #pragma unroll
    for (int t = 0; t < 8; ++t) {
        v2f a; a.x = arl[2*t]; a.y = arl[2*t+1];
        int kl = 4 * t + ks;                                // even
        v2f b1 = *(const v2f*)&sWst[(kl >> 1) * 256 + col * 2];
        v2f b2 = *(const v2f*)&sWts[(kl >> 1) * 256 + col * 2];
        acc_st = __builtin_amdgcn_wmma_f32_16x16x4_f32(false, a, false, b1, (short)0, acc_st, false, false);
        acc_ts = __builtin_amdgcn_wmma_f32_16x16x4_f32(false, a, false, b2, (short)0, acc_ts, false, false);
    }
    int orow0 = rowBase + hi * 8;
    #pragma unroll
    for (int j = 0; j < 8; ++j) {
        int rr = orow0 + j;
        if (rr < E) {
            Out[(size_t)rr * D_EDGE + col] = INV_SQRT_2 * silu(acc_st[j]);
            Z[(size_t)rr * D_EDGE + col]   = silu(acc_ts[j]);
        }
    }
}

// ---------------- pass 6: out[e] += c * z[idx_swap[e]]  (Z read is last-use -> NT) -------
__global__ void k_swap_add(const float* __restrict__ Z, const int* __restrict__ sw,
                           float* __restrict__ Out, int E) {
    int i = blockIdx.x * blockDim.x + threadIdx.x;
    int n = E * D_EDGE;
    if (i < n) {
        int e = i >> 7, j = i & 127;
        int se = sw[e];
        if ((unsigned)se < (unsigned)E) {
            float zv = __builtin_nontemporal_load(Z + (size_t)se * D_EDGE + j);
            Out[i] += INV_SQRT_2 * zv;
        }
    }
}

extern "C" void kernel_launch(void* const* d_in, const int* in_sizes, int n_in,
                              void* d_out, int out_size, void* d_ws, size_t ws_size,
                              hipStream_t stream) {
    (void)n_in; (void)out_size; (void)ws_size;
    const float* m_st        = (const float*)d_in[0];
    const float* sbf         = (const float*)d_in[1];
    const int*   idx_swap    = (const int*)  d_in[2];
    const int*   edge_nb_idx = (const int*)  d_in[3];
    const int*   edge_nb_rag = (const int*)  d_in[4];
    const float* W_down      = (const float*)d_in[5];
    const float* W_bil       = (const float*)d_in[6];
    const float* W_up_st     = (const float*)d_in[7];
    const float* W_up_ts     = (const float*)d_in[8];
    const float* scale_sbf   = (const float*)d_in[9];

    int E    = in_sizes[0] / D_EDGE;
    int E_NB = in_sizes[1] / D_SBF;

    float* ws   = (float*)d_ws;
    float* Mbuf = ws;                                     // E*32
    float* Sbar = ws + (size_t)E * D_QUAD;                // E*32
    float* Xbuf = ws + 2 * (size_t)E * D_QUAD;            // E*32
    float* Zbuf = ws + 3 * (size_t)E * D_QUAD;            // E*128
    int*   inv  = (int*)(ws + 3 * (size_t)E * D_QUAD + (size_t)E * D_EDGE); // E*16 ints

    int nInv = E * KMAX;
    k_init_inv   <<<(nInv + 255) / 256, 256, 0, stream>>>(inv, nInv);
    k_scatter_inv<<<(E_NB + 255) / 256, 256, 0, stream>>>(edge_nb_idx, edge_nb_rag, inv, E, E_NB);
    k_gemm_down  <<<(E + 31) / 32, 128, 0, stream>>>(m_st, W_down, Mbuf, E);
    k_reduce_sbf <<<(E + 31) / 32, 256, 0, stream>>>(sbf, inv, Sbar, E);
    k_bilinear   <<<(E + 63) / 64, 256, 1024 * 32 * sizeof(float), stream>>>(Mbuf, Sbar, W_bil, scale_sbf, Xbuf, E);
    k_gemm_up    <<<(E + 15) / 16, 256, 0, stream>>>(Xbuf, W_up_st, W_up_ts, (float*)d_out, Zbuf, E);
    int nOut = E * D_EDGE;
    k_swap_add   <<<(nOut + 255) / 256, 256, 0, stream>>>(Zbuf, idx_swap, (float*)d_out, E);
}